// SMPL_19327352832329
// MI455X (gfx1250) — compile-verified
//
#include <hip/hip_runtime.h>
#include <cstddef>

#define B_   512
#define V_   6890
#define NB_  10
#define NJ_  24
#define NK_  14
#define K3   20670   // V_*3
#define NPAD 20672   // padded column count (multiple of 64)
#define KD   217     // NB_ + 207
#define KPAD 224
#define KTILES 55    // 55*4 = 220 >= 217
#define NT4  323     // 20672/64 column-tiles of width 64
#define MTILES 32    // 512/16

typedef __attribute__((ext_vector_type(2))) float v2f;
typedef __attribute__((ext_vector_type(8))) float v8f;

// workspace layout (float offsets)
static constexpr size_t O_RS = 0;                            // B*216    Rs
static constexpr size_t O_G  = O_RS + (size_t)B_ * 216;      // B*224    [beta|pf|0]
static constexpr size_t O_SJ = O_G  + (size_t)B_ * KPAD;     // 11*72    SJ + Jt
static constexpr size_t O_J  = O_SJ + 792;                   // B*72     joints J
static constexpr size_t O_RR = O_J  + (size_t)B_ * 72;       // B*288    chain results
static constexpr size_t O_A  = O_RR + (size_t)B_ * 288;      // B*288    A (3x4 each)
static constexpr size_t O_W  = O_A  + (size_t)B_ * 288;      // 224*20672 packed+padded [sd;pd]
static constexpr size_t O_VP = O_W  + (size_t)KPAD * NPAD;   // B*20672  v_posed -> verts
// total ~63.1 MB

__global__ void k_rodrigues(const float* __restrict__ param, float* __restrict__ ws) {
  int t = blockIdx.x * blockDim.x + threadIdx.x;
  if (t >= B_ * NJ_) return;
  int b = t / NJ_, j = t % NJ_;
  const float* th = param + (size_t)b * (NJ_ * 3) + j * 3;
  float x = th[0], y = th[1], z = th[2];
  float ex = x + 1e-8f, ey = y + 1e-8f, ez = z + 1e-8f;
  float angle = sqrtf(ex * ex + ey * ey + ez * ez);
  float rx = x / angle, ry = y / angle, rz = z / angle;
  float c = cosf(angle), s = sinf(angle), o = 1.0f - c;
  float* R = ws + O_RS + (size_t)b * 216 + j * 9;
  R[0] = c + o * rx * rx;      R[1] = o * rx * ry - s * rz; R[2] = o * rx * rz + s * ry;
  R[3] = o * ry * rx + s * rz; R[4] = c + o * ry * ry;      R[5] = o * ry * rz - s * rx;
  R[6] = o * rz * rx - s * ry; R[7] = o * rz * ry + s * rx; R[8] = c + o * rz * rz;
}

__global__ void k_buildG(const float* __restrict__ beta, float* __restrict__ ws) {
  int t = blockIdx.x * blockDim.x + threadIdx.x;
  if (t >= B_ * KPAD) return;
  int b = t / KPAD, col = t % KPAD;
  float v = 0.0f;
  if (col < NB_) {
    v = beta[b * NB_ + col];
  } else if (col < KD) {
    int q = col - NB_;
    int j = q / 9 + 1;
    int rc = q % 9;
    v = ws[O_RS + (size_t)b * 216 + j * 9 + rc] - ((rc == 0 || rc == 4 || rc == 8) ? 1.0f : 0.0f);
  }
  ws[O_G + (size_t)b * KPAD + col] = v;
}

// Pack W = [shapedirs ; posedirs ; zeros] into (KPAD x NPAD), zero-padded rows/cols.
__global__ void k_buildW(const float* __restrict__ sd, const float* __restrict__ pd,
                         float* __restrict__ ws) {
  int col = blockIdx.x * blockDim.x + threadIdx.x;
  int kr  = blockIdx.y;
  if (col >= NPAD) return;
  float v = 0.0f;
  if (col < K3) {
    if (kr < NB_)      v = sd[(size_t)kr * K3 + col];
    else if (kr < KD)  v = pd[(size_t)(kr - NB_) * K3 + col];
  }
  ws[O_W + (size_t)kr * NPAD + col] = v;
}

// SJ[n][j][c] for n<10; n==10 is the v_template term Jt[j][c]
__global__ void k_SJ(const float* __restrict__ sd, const float* __restrict__ vt,
                     const float* __restrict__ Jreg, float* __restrict__ ws) {
  int n = blockIdx.x / 72;
  int jc = blockIdx.x % 72;
  int j = jc / 3, c = jc % 3;
  float p = 0.0f;
  for (int v = threadIdx.x; v < V_; v += 256) {
    float x = (n < NB_) ? sd[(size_t)n * K3 + v * 3 + c] : vt[v * 3 + c];
    p += x * Jreg[v * NJ_ + j];
  }
  __shared__ float red[256];
  red[threadIdx.x] = p;
  __syncthreads();
  for (int s = 128; s > 0; s >>= 1) {
    if (threadIdx.x < s) red[threadIdx.x] += red[threadIdx.x + s];
    __syncthreads();
  }
  if (threadIdx.x == 0) ws[O_SJ + n * 72 + jc] = red[0];
}

__global__ void k_J(const float* __restrict__ beta, float* __restrict__ ws) {
  int t = blockIdx.x * blockDim.x + threadIdx.x;
  if (t >= B_ * 72) return;
  int b = t / 72, jc = t % 72;
  float v = ws[O_SJ + 10 * 72 + jc];
  for (int n = 0; n < NB_; ++n) v += beta[b * NB_ + n] * ws[O_SJ + n * 72 + jc];
  ws[O_J + (size_t)b * 72 + jc] = v;
}

__global__ void k_chain(float* __restrict__ ws) {
  int b = blockIdx.x * blockDim.x + threadIdx.x;
  if (b >= B_) return;
  const int PAR[NJ_] = {-1, 0, 0, 0, 1, 2, 3, 4, 5, 6, 7, 8, 9, 9, 9, 12, 13, 14, 16, 17, 18, 19, 20, 21};
  const float* Rsb = ws + O_RS + (size_t)b * 216;
  const float* Jb  = ws + O_J  + (size_t)b * 72;
  float* RR = ws + O_RR + (size_t)b * 288;
  float* Ab = ws + O_A  + (size_t)b * 288;
  for (int i = 0; i < NJ_; ++i) {
    const float* R = Rsb + i * 9;
    float out[12];
    if (i == 0) {
      for (int r = 0; r < 3; ++r) {
        out[r * 4 + 0] = R[r * 3 + 0];
        out[r * 4 + 1] = R[r * 3 + 1];
        out[r * 4 + 2] = R[r * 3 + 2];
        out[r * 4 + 3] = Jb[r];
      }
    } else {
      int p = PAR[i];
      float tx = Jb[i * 3 + 0] - Jb[p * 3 + 0];
      float ty = Jb[i * 3 + 1] - Jb[p * 3 + 1];
      float tz = Jb[i * 3 + 2] - Jb[p * 3 + 2];
      const float* P = RR + p * 12;
      for (int r = 0; r < 3; ++r) {
        for (int c = 0; c < 3; ++c)
          out[r * 4 + c] = P[r * 4 + 0] * R[0 * 3 + c] + P[r * 4 + 1] * R[1 * 3 + c] + P[r * 4 + 2] * R[2 * 3 + c];
        out[r * 4 + 3] = P[r * 4 + 0] * tx + P[r * 4 + 1] * ty + P[r * 4 + 2] * tz + P[r * 4 + 3];
      }
    }
    for (int e = 0; e < 12; ++e) RR[i * 12 + e] = out[e];
    float jx = Jb[i * 3], jy = Jb[i * 3 + 1], jz = Jb[i * 3 + 2];
    for (int r = 0; r < 3; ++r) {
      float ib = out[r * 4] * jx + out[r * 4 + 1] * jy + out[r * 4 + 2] * jz;
      Ab[i * 12 + r * 4 + 0] = out[r * 4 + 0];
      Ab[i * 12 + r * 4 + 1] = out[r * 4 + 1];
      Ab[i * 12 + r * 4 + 2] = out[r * 4 + 2];
      Ab[i * 12 + r * 4 + 3] = out[r * 4 + 3] - ib;
    }
  }
}

// VP(512 x NPAD) = G(512 x 220) @ W(220 x NPAD) + v_template (per column).
// One wave per 16(M) x 64(N) tile: A-frag shared across 4 accumulators.
__global__ __launch_bounds__(32) void k_gemm(float* __restrict__ ws,
                                             const float* __restrict__ vt) {
  const float* G = ws + O_G;
  const float* W = ws + O_W;
  float* VP = ws + O_VP;
  int lane = threadIdx.x;
  int hi = lane >> 4, lo = lane & 15;
  int col0 = blockIdx.x * 64 + lo;     // base column of this wave's 4 N-tiles
  int mtile = blockIdx.y;
  int arow = mtile * 16 + lo;          // A fragment: M = lane&15

  v8f acc0 = {}, acc1 = {}, acc2 = {}, acc3 = {};
#pragma unroll 5
  for (int kk = 0; kk < KTILES; ++kk) {
    int k0 = kk * 4 + hi * 2;          // lanes 16-31 carry K+2, K+3
    v2f a;
    a.x = G[(size_t)arow * KPAD + k0];
    a.y = G[(size_t)arow * KPAD + k0 + 1];
    const float* Wr0 = W + (size_t)k0 * NPAD + col0;
    const float* Wr1 = Wr0 + NPAD;
    v2f b0, b1, b2, b3;
    b0.x = Wr0[0];  b0.y = Wr1[0];
    b1.x = Wr0[16]; b1.y = Wr1[16];
    b2.x = Wr0[32]; b2.y = Wr1[32];
    b3.x = Wr0[48]; b3.y = Wr1[48];
    acc0 = __builtin_amdgcn_wmma_f32_16x16x4_f32(false, a, false, b0, (short)0, acc0, false, false);
    acc1 = __builtin_amdgcn_wmma_f32_16x16x4_f32(false, a, false, b1, (short)0, acc1, false, false);
    acc2 = __builtin_amdgcn_wmma_f32_16x16x4_f32(false, a, false, b2, (short)0, acc2, false, false);
    acc3 = __builtin_amdgcn_wmma_f32_16x16x4_f32(false, a, false, b3, (short)0, acc3, false, false);
  }
  int rbase = mtile * 16 + hi * 8;     // C: VGPR i holds rows i (lanes 0-15) / i+8 (lanes 16-31)
  v8f accs[4] = {acc0, acc1, acc2, acc3};
#pragma unroll
  for (int t = 0; t < 4; ++t) {
    int col = col0 + 16 * t;
    int cc = col < K3 ? col : (K3 - 1);  // clamp: padded-column values are never consumed
    float tadd = vt[cc];
#pragma unroll
    for (int i = 0; i < 8; ++i)
      VP[(size_t)(rbase + i) * NPAD + col] = accs[t][i] + tadd;
  }
}

// Fused skinning: T = sum_j w[v,j]*A[b,j]; verts = T * (vp,1); write verts over VP in place.
__global__ void k_skin(const float* __restrict__ lbsw, float* __restrict__ ws) {
  __shared__ float sA[288];
  int b = blockIdx.y;
  for (int e = threadIdx.x; e < 288; e += blockDim.x) sA[e] = ws[O_A + (size_t)b * 288 + e];
  __syncthreads();
  int v = blockIdx.x * blockDim.x + threadIdx.x;
  if (v >= V_) return;
  float T[12];
#pragma unroll
  for (int e = 0; e < 12; ++e) T[e] = 0.0f;
  for (int j = 0; j < NJ_; ++j) {
    float w = lbsw[(size_t)v * NJ_ + j];
#pragma unroll
    for (int e = 0; e < 12; ++e) T[e] += w * sA[j * 12 + e];
  }
  float* vp = ws + O_VP + (size_t)b * NPAD + v * 3;
  float x = vp[0], y = vp[1], z = vp[2];
  vp[0] = T[0] * x + T[1] * y + T[2]  * z + T[3];
  vp[1] = T[4] * x + T[5] * y + T[6]  * z + T[7];
  vp[2] = T[8] * x + T[9] * y + T[10] * z + T[11];
}

__global__ void k_joints(const float* __restrict__ jr, const float* __restrict__ ws,
                         float* __restrict__ out) {
  int b = blockIdx.y;
  int k = blockIdx.x / 3, c = blockIdx.x % 3;
  float p = 0.0f;
  for (int v = threadIdx.x; v < V_; v += 256)
    p += ws[O_VP + (size_t)b * NPAD + v * 3 + c] * jr[v * NK_ + k];
  __shared__ float red[256];
  red[threadIdx.x] = p;
  __syncthreads();
  for (int s = 128; s > 0; s >>= 1) {
    if (threadIdx.x < s) red[threadIdx.x] += red[threadIdx.x + s];
    __syncthreads();
  }
  if (threadIdx.x == 0) out[(size_t)b * (NK_ * 3) + k * 3 + c] = red[0];
}

extern "C" void kernel_launch(void* const* d_in, const int* in_sizes, int n_in,
                              void* d_out, int out_size, void* d_ws, size_t ws_size,
                              hipStream_t stream) {
  (void)in_sizes; (void)n_in; (void)out_size; (void)ws_size;
  const float* beta  = (const float*)d_in[0];
  const float* param = (const float*)d_in[1];
  const float* vt    = (const float*)d_in[2];
  const float* sd    = (const float*)d_in[3];
  const float* Jreg  = (const float*)d_in[4];
  const float* pd    = (const float*)d_in[5];
  const float* lbsw  = (const float*)d_in[6];
  const float* jr    = (const float*)d_in[7];
  float* out = (float*)d_out;
  float* ws  = (float*)d_ws;

  k_rodrigues<<<(B_ * NJ_ + 255) / 256, 256, 0, stream>>>(param, ws);
  k_buildG<<<(B_ * KPAD + 255) / 256, 256, 0, stream>>>(beta, ws);
  k_buildW<<<dim3((NPAD + 255) / 256, KPAD), 256, 0, stream>>>(sd, pd, ws);
  k_SJ<<<11 * 72, 256, 0, stream>>>(sd, vt, Jreg, ws);
  k_J<<<(B_ * 72 + 255) / 256, 256, 0, stream>>>(beta, ws);
  k_chain<<<(B_ + 255) / 256, 256, 0, stream>>>(ws);
  k_gemm<<<dim3(NT4, MTILES), 32, 0, stream>>>(ws, vt);
  k_skin<<<dim3((V_ + 255) / 256, B_), 256, 0, stream>>>(lbsw, ws);
  k_joints<<<dim3(NK_ * 3, B_), 256, 0, stream>>>(jr, ws, out);
}